// GNN_17025250361854
// MI455X (gfx1250) — compile-verified
//
#include <hip/hip_runtime.h>
#include <math.h>

// ---------------------------------------------------------------------------
// 2-layer GCN (PyG GCNConv semantics) for MI455X / gfx1250.
// Dense 16-row tiles use V_WMMA_F32_16X16X4_F32 (f32 WMMA, wave32), with the
// self-loop + bias epilogue fused into the GEMM (accumulator is already in
// VGPRs, so no re-read of h1/h2). Edge gather/scatter uses global f32 atomics;
// feature matrices are compact N x 16 rows (32 MB each) so h and agg both stay
// resident in the 192 MB L2 — the scatter runs at L2, not HBM, bandwidth.
// ---------------------------------------------------------------------------

typedef __attribute__((ext_vector_type(2))) float v2f;
typedef __attribute__((ext_vector_type(8))) float v8f;

// ---------------- degree / normalization -----------------------------------

__global__ void k_init_deg(float* __restrict__ deg, int n) {
  int i = blockIdx.x * blockDim.x + threadIdx.x;
  if (i < n) deg[i] = 1.0f;  // self-loop contribution
}

__global__ void k_count_deg(const int* __restrict__ dst, float* __restrict__ deg, int e) {
  int i = blockIdx.x * blockDim.x + threadIdx.x;
  if (i < e) atomicAdd(&deg[dst[i]], 1.0f);
}

__global__ void k_dinv(float* __restrict__ deg, int n) {
  int i = blockIdx.x * blockDim.x + threadIdx.x;
  if (i < n) deg[i] = rsqrtf(deg[i]);  // deg >= 1 always (self-loop)
}

// ---------------- GEMM1: h1 = x @ W1  (N x 3) @ (3 x 16) --------------------
// One wave per 16-row tile; single V_WMMA_F32_16X16X4_F32 with K padded 3->4.
// A layout (16x4 f32): lanes 0-15 hold {K=0,K=1}, lanes 16-31 hold {K=2,K=3},
// row M = lane & 15. B layout (4x16): N = lane & 15, same K split.
// Fused epilogue: agg1 = dinv^2 * h1 + b1 (self-loop term + bias).
__global__ void k_gemm1_wmma(const float* __restrict__ x,
                             const float* __restrict__ W1,
                             const float* __restrict__ dinv,
                             const float* __restrict__ b1,
                             float* __restrict__ h1,
                             float* __restrict__ agg1, int n) {
  int lane = threadIdx.x & 31;
  int wave = (blockIdx.x * blockDim.x + threadIdx.x) >> 5;
  int row0 = wave * 16;
  if (row0 >= n) return;  // wave-uniform: EXEC stays all-ones for WMMA

  int half = lane >> 4;
  int r    = lane & 15;
  int row  = row0 + r;

  v2f a, b;
  if (half == 0) {
    a.x = (row < n) ? x[row * 3 + 0] : 0.0f;
    a.y = (row < n) ? x[row * 3 + 1] : 0.0f;
    b.x = W1[0 * 16 + r];
    b.y = W1[1 * 16 + r];
  } else {
    a.x = (row < n) ? x[row * 3 + 2] : 0.0f;
    a.y = 0.0f;                 // padded K=3
    b.x = W1[2 * 16 + r];
    b.y = 0.0f;                 // padded K=3
  }

  v8f c = {};
  c = __builtin_amdgcn_wmma_f32_16x16x4_f32(false, a, false, b,
                                            (short)0, c, false, false);

  float bias = b1[r];
  // D layout: VGPR v holds M = v + 8*half, N = r
#pragma unroll
  for (int v = 0; v < 8; ++v) {
    int m = row0 + v + 8 * half;
    float di = dinv[m];                 // uniform across the 16 N-lanes
    float hv = c[v];
    h1[m * 16 + r]   = hv;
    agg1[m * 16 + r] = di * di * hv + bias;
  }
}

// ---------------- edge scatter layer 1 (16 features) ------------------------

__global__ void k_scatter1(const int* __restrict__ src, const int* __restrict__ dst,
                           const float* __restrict__ dinv,
                           const float* __restrict__ h1,
                           float* __restrict__ agg1, int e) {
  int i = blockIdx.x * blockDim.x + threadIdx.x;
  if (i >= e) return;
  int s = src[i], d = dst[i];
  float w = dinv[s] * dinv[d];
  const float4* hs = (const float4*)(h1 + (long long)s * 16);
  float4 v0 = hs[0], v1 = hs[1], v2 = hs[2], v3 = hs[3];
  float* o = agg1 + (long long)d * 16;
  atomicAdd(o + 0,  w * v0.x); atomicAdd(o + 1,  w * v0.y);
  atomicAdd(o + 2,  w * v0.z); atomicAdd(o + 3,  w * v0.w);
  atomicAdd(o + 4,  w * v1.x); atomicAdd(o + 5,  w * v1.y);
  atomicAdd(o + 6,  w * v1.z); atomicAdd(o + 7,  w * v1.w);
  atomicAdd(o + 8,  w * v2.x); atomicAdd(o + 9,  w * v2.y);
  atomicAdd(o + 10, w * v2.z); atomicAdd(o + 11, w * v2.w);
  atomicAdd(o + 12, w * v3.x); atomicAdd(o + 13, w * v3.y);
  atomicAdd(o + 14, w * v3.z); atomicAdd(o + 15, w * v3.w);
}

// ---------------- GEMM2: h2 = relu(agg1) @ W2  (N x 16) @ (16 x 10) ---------
// Four chained K=4 WMMAs; ReLU fused into the A-operand load; W2 padded to
// 16 columns (cols 10..15 = 0). Fused epilogue writes h2 (row stride 16, for
// the edge gather) AND initializes out = dinv^2 * h2 + b2 (row stride 10).
__global__ void k_gemm2_wmma(const float* __restrict__ agg1,
                             const float* __restrict__ W2,
                             const float* __restrict__ dinv,
                             const float* __restrict__ b2,
                             float* __restrict__ h2,
                             float* __restrict__ out, int n) {
  int lane = threadIdx.x & 31;
  int wave = (blockIdx.x * blockDim.x + threadIdx.x) >> 5;
  int row0 = wave * 16;
  if (row0 >= n) return;  // wave-uniform

  int half = lane >> 4;
  int r    = lane & 15;
  int row  = row0 + r;
  bool cok = (r < 10);

  v8f c = {};
#pragma unroll
  for (int k = 0; k < 4; ++k) {
    int ka = 4 * k + 2 * half;  // first of the two K rows this half-wave owns
    v2f a, b;
    a.x = (row < n) ? fmaxf(agg1[(long long)row * 16 + ka + 0], 0.0f) : 0.0f;
    a.y = (row < n) ? fmaxf(agg1[(long long)row * 16 + ka + 1], 0.0f) : 0.0f;
    b.x = cok ? W2[(ka + 0) * 10 + r] : 0.0f;   // B(K=ka,   N=r)
    b.y = cok ? W2[(ka + 1) * 10 + r] : 0.0f;   // B(K=ka+1, N=r)
    c = __builtin_amdgcn_wmma_f32_16x16x4_f32(false, a, false, b,
                                              (short)0, c, false, false);
  }

  float bias = cok ? b2[r] : 0.0f;
#pragma unroll
  for (int v = 0; v < 8; ++v) {
    int m = row0 + v + 8 * half;
    float di = dinv[m];
    float hv = c[v];
    h2[m * 16 + r] = hv;                       // gather source for scatter2
    if (cok) out[m * 10 + r] = di * di * hv + bias;  // self-loop + bias init
  }
}

// ---------------- edge scatter layer 2 (10 features) ------------------------

__global__ void k_scatter2(const int* __restrict__ src, const int* __restrict__ dst,
                           const float* __restrict__ dinv,
                           const float* __restrict__ h2,
                           float* __restrict__ out, int e) {
  int i = blockIdx.x * blockDim.x + threadIdx.x;
  if (i >= e) return;
  int s = src[i], d = dst[i];
  float w = dinv[s] * dinv[d];
  const float* hs = h2 + (long long)s * 16;
  float* o = out + (long long)d * 10;
#pragma unroll
  for (int cc = 0; cc < 10; ++cc) atomicAdd(o + cc, w * hs[cc]);
}

// ---------------- log-softmax over 10 classes, in place ---------------------

__global__ void k_logsoftmax(float* __restrict__ out, int n) {
  int i = blockIdx.x * blockDim.x + threadIdx.x;
  if (i >= n) return;
  float* row = out + (long long)i * 10;
  float v[10];
  float m = -INFINITY;
#pragma unroll
  for (int cc = 0; cc < 10; ++cc) { v[cc] = row[cc]; m = fmaxf(m, v[cc]); }
  float s = 0.0f;
#pragma unroll
  for (int cc = 0; cc < 10; ++cc) s += __expf(v[cc] - m);
  float lse = m + __logf(s);
#pragma unroll
  for (int cc = 0; cc < 10; ++cc) row[cc] = v[cc] - lse;
}

// ---------------------------------------------------------------------------

extern "C" void kernel_launch(void* const* d_in, const int* in_sizes, int n_in,
                              void* d_out, int out_size, void* d_ws, size_t ws_size,
                              hipStream_t stream) {
  const float* x  = (const float*)d_in[0];   // [N,3]
  const int*   ei = (const int*)d_in[1];     // [2,E]
  const float* W1 = (const float*)d_in[2];   // [3,16]
  const float* b1 = (const float*)d_in[3];   // [16]
  const float* W2 = (const float*)d_in[4];   // [16,10]
  const float* b2 = (const float*)d_in[5];   // [10]
  float* out = (float*)d_out;                // [N,10]

  const int N = in_sizes[0] / 3;
  const int E = in_sizes[1] / 2;
  const int* src = ei;
  const int* dst = ei + E;

  // Workspace layout (floats): dinv[N] | h1[16N] | agg1[16N] | h2[16N]
  float* ws   = (float*)d_ws;
  float* dinv = ws;
  float* h1   = ws + (size_t)N;
  float* agg1 = ws + (size_t)N * 17;
  float* h2   = ws + (size_t)N * 33;

  const int T = 256;
  int tiles = (N + 15) / 16;                 // 16-row WMMA tiles
  int gemm_blocks = (tiles + (T / 32) - 1) / (T / 32);

  // normalization: deg = 1 + in-degree; dinv = rsqrt(deg)
  k_init_deg <<<(N + T - 1) / T, T, 0, stream>>>(dinv, N);
  k_count_deg<<<(E + T - 1) / T, T, 0, stream>>>(dst, dinv, E);
  k_dinv     <<<(N + T - 1) / T, T, 0, stream>>>(dinv, N);

  // layer 1: WMMA GEMM with fused self-loop+bias epilogue, then edge scatter
  k_gemm1_wmma<<<gemm_blocks, T, 0, stream>>>(x, W1, dinv, b1, h1, agg1, N);
  k_scatter1  <<<(E + T - 1) / T, T, 0, stream>>>(src, dst, dinv, h1, agg1, E);

  // layer 2: ReLU fused into A-load, out-init fused into epilogue, scatter
  k_gemm2_wmma<<<gemm_blocks, T, 0, stream>>>(agg1, W2, dinv, b2, h2, out, N);
  k_scatter2  <<<(E + T - 1) / T, T, 0, stream>>>(src, dst, dinv, h2, out, E);

  // log_softmax in place on d_out
  k_logsoftmax<<<(N + T - 1) / T, T, 0, stream>>>(out, N);
}